// MultiheadSelfAttention_19104014533004
// MI455X (gfx1250) — compile-verified
//
#include <hip/hip_runtime.h>

typedef __attribute__((ext_vector_type(2))) float v2f;
typedef __attribute__((ext_vector_type(8))) float v8f;

#define WMMA_F32(a, b, c) \
  __builtin_amdgcn_wmma_f32_16x16x4_f32(false, (a), false, (b), (short)0, (c), false, false)

// Problem constants
#define BATCH 4
#define SEQ   2048
#define KDIM  1024
#define HEADS 16
#define HS    64              // head size
#define ROWS  (BATCH * SEQ)   // 8192

__device__ __forceinline__ float rowmax16(float x) {
  x = fmaxf(x, __shfl_xor(x, 1, 32));
  x = fmaxf(x, __shfl_xor(x, 2, 32));
  x = fmaxf(x, __shfl_xor(x, 4, 32));
  x = fmaxf(x, __shfl_xor(x, 8, 32));
  return x;
}
__device__ __forceinline__ float rowsum16(float x) {
  x += __shfl_xor(x, 1, 32);
  x += __shfl_xor(x, 2, 32);
  x += __shfl_xor(x, 4, 32);
  x += __shfl_xor(x, 8, 32);
  return x;
}

// ---------------------------------------------------------------------------
// Kernel 1: fused Q/K/V projection, 64x64 output tile per wave.
// y = x @ W^T : A[m][k] and B[k][n] = W[n][k]; each lane's K-adjacent fp32
// pair is contiguous -> b64 fragment loads.  4 A-frags + 4 B-frags per K-step
// feed 16 WMMAs (16 flops/byte of L2 traffic).
// ---------------------------------------------------------------------------
__global__ void __launch_bounds__(32)
qkv_proj_kernel(const float* __restrict__ x,
                const float* __restrict__ Wq, const float* __restrict__ Wk,
                const float* __restrict__ Wv,
                float* __restrict__ qo, float* __restrict__ ko,
                float* __restrict__ vo) {
  const int lane = threadIdx.x;
  const int half = lane >> 4;
  const int l16  = lane & 15;
  const int m0 = blockIdx.x * 64;   // row tile
  const int n0 = blockIdx.y * 64;   // col tile
  const int which = blockIdx.z;

  const float* W  = (which == 0) ? Wq : (which == 1) ? Wk : Wv;
  float*      out = (which == 0) ? qo : (which == 1) ? ko : vo;

  v8f acc[4][4];
#pragma unroll
  for (int i = 0; i < 4; ++i)
#pragma unroll
    for (int j = 0; j < 4; ++j)
#pragma unroll
      for (int r = 0; r < 8; ++r) acc[i][j][r] = 0.0f;

  const float* arow[4];
  const float* brow[4];
#pragma unroll
  for (int i = 0; i < 4; ++i) {
    arow[i] = x + (size_t)(m0 + i * 16 + l16) * KDIM;
    brow[i] = W + (size_t)(n0 + i * 16 + l16) * KDIM;
  }

  for (int kk = 0; kk < KDIM; kk += 4) {
    const int ka = kk + 2 * half;
    v2f a[4], b[4];
#pragma unroll
    for (int i = 0; i < 4; ++i) a[i] = *(const v2f*)(arow[i] + ka);
#pragma unroll
    for (int j = 0; j < 4; ++j) b[j] = *(const v2f*)(brow[j] + ka);
#pragma unroll
    for (int i = 0; i < 4; ++i)
#pragma unroll
      for (int j = 0; j < 4; ++j)
        acc[i][j] = WMMA_F32(a[i], b[j], acc[i][j]);
  }

#pragma unroll
  for (int i = 0; i < 4; ++i)
#pragma unroll
    for (int j = 0; j < 4; ++j)
#pragma unroll
      for (int r = 0; r < 8; ++r)
        out[(size_t)(m0 + i * 16 + r + 8 * half) * KDIM + n0 + j * 16 + l16] =
            acc[i][j][r];
}

// ---------------------------------------------------------------------------
// Kernel 2: flash attention, fp32 WMMA.  One wave handles a 32-query tile for
// one (batch, head): K- and V-fragments are loaded once per key tile and
// reused by two 16-query score/output tiles (16 flops/byte).
// K/V slice per (b,h) is 1 MB -> L2-resident.
// ---------------------------------------------------------------------------
__global__ void __launch_bounds__(32)
flash_attn_kernel(const float* __restrict__ q, const float* __restrict__ k,
                  const float* __restrict__ v, float* __restrict__ o) {
  __shared__ float p_lds[32 * 18];  // stride 18: conflict-free, 8B-aligned pairs

  const int lane = threadIdx.x;
  const int half = lane >> 4;
  const int l16  = lane & 15;
  const int t0 = blockIdx.x * 32;   // query tile (32 rows)
  const int h  = blockIdx.y;        // head
  const int b  = blockIdx.z;        // batch

  const float* qbase = q + ((size_t)b * SEQ + t0) * KDIM + h * HS;
  const float* kbase = k + (size_t)b * SEQ * KDIM + h * HS;
  const float* vbase = v + (size_t)b * SEQ * KDIM + h * HS;

  // fold softmax scale (1/sqrt(K)) and log2(e) into q -> exp2 softmax
  const float scale = 0.03125f * 1.44269504088896340736f;

  // Q tile as 2 x 16 A-fragments (K=4 each over S=64), pre-scaled
  v2f qa[2][16];
#pragma unroll
  for (int u = 0; u < 2; ++u)
#pragma unroll
    for (int kk = 0; kk < 16; ++kk) {
      v2f t = *(const v2f*)(qbase + (size_t)(u * 16 + l16) * KDIM + kk * 4 + 2 * half);
      qa[u][kk].x = t.x * scale;
      qa[u][kk].y = t.y * scale;
    }

  float m_r[2][8], l_r[2][8];
  v8f oacc[2][4];
#pragma unroll
  for (int u = 0; u < 2; ++u) {
#pragma unroll
    for (int r = 0; r < 8; ++r) { m_r[u][r] = -3.0e38f; l_r[u][r] = 0.0f; }
#pragma unroll
    for (int j = 0; j < 4; ++j)
#pragma unroll
      for (int r = 0; r < 8; ++r) oacc[u][j][r] = 0.0f;
  }

  for (int kt = 0; kt < SEQ; kt += 16) {
    // ---- scores S[u] = (q*scale) @ k^T : k-fragment shared by both q tiles ----
    v8f s[2];
#pragma unroll
    for (int u = 0; u < 2; ++u)
#pragma unroll
      for (int r = 0; r < 8; ++r) s[u][r] = 0.0f;
#pragma unroll
    for (int kk = 0; kk < 16; ++kk) {
      v2f bfrag = *(const v2f*)(kbase + (size_t)(kt + l16) * KDIM + kk * 4 + 2 * half);
      s[0] = WMMA_F32(qa[0][kk], bfrag, s[0]);
      s[1] = WMMA_F32(qa[1][kk], bfrag, s[1]);
    }

    // ---- online softmax (row = u*16 + r + 8*half, col = l16) ----
#pragma unroll
    for (int u = 0; u < 2; ++u)
#pragma unroll
      for (int r = 0; r < 8; ++r) {
        float tmax = rowmax16(s[u][r]);
        float mn   = fmaxf(m_r[u][r], tmax);
        float corr = exp2f(m_r[u][r] - mn);
        float p    = exp2f(s[u][r] - mn);
        l_r[u][r] = l_r[u][r] * corr + rowsum16(p);
        m_r[u][r] = mn;
#pragma unroll
        for (int j = 0; j < 4; ++j) oacc[u][j][r] *= corr;
        p_lds[(u * 16 + r + 8 * half) * 18 + l16] = p;  // D layout -> LDS
      }
    __syncthreads();  // 1-wave WG: S_NOP, compiler fence for LDS relayout

    // ---- O[u] += P[u] @ V : v-fragment shared by both q tiles ----
#pragma unroll
    for (int kk = 0; kk < 4; ++kk) {
      const int key = kk * 4 + 2 * half;
      v2f pa0 = *(const v2f*)(&p_lds[(l16) * 18 + key]);
      v2f pa1 = *(const v2f*)(&p_lds[(16 + l16) * 18 + key]);
#pragma unroll
      for (int j = 0; j < 4; ++j) {
        v2f vb;
        vb.x = vbase[(size_t)(kt + key) * KDIM + j * 16 + l16];
        vb.y = vbase[(size_t)(kt + key + 1) * KDIM + j * 16 + l16];
        oacc[0][j] = WMMA_F32(pa0, vb, oacc[0][j]);
        oacc[1][j] = WMMA_F32(pa1, vb, oacc[1][j]);
      }
    }
    __syncthreads();
  }

  // ---- normalize and store ----
#pragma unroll
  for (int u = 0; u < 2; ++u)
#pragma unroll
    for (int r = 0; r < 8; ++r) {
      const float inv = 1.0f / l_r[u][r];
#pragma unroll
      for (int j = 0; j < 4; ++j)
        o[((size_t)b * SEQ + t0 + u * 16 + r + 8 * half) * KDIM + h * HS +
          j * 16 + l16] = oacc[u][j][r] * inv;
    }
}

// ---------------------------------------------------------------------------
// Kernel 3: output projection  out = o @ Wu^T + bu, 64x64 tile per wave
// ---------------------------------------------------------------------------
__global__ void __launch_bounds__(32)
out_proj_kernel(const float* __restrict__ A, const float* __restrict__ Wu,
                const float* __restrict__ bu, float* __restrict__ out) {
  const int lane = threadIdx.x;
  const int half = lane >> 4;
  const int l16  = lane & 15;
  const int m0 = blockIdx.x * 64;
  const int n0 = blockIdx.y * 64;

  v8f acc[4][4];
#pragma unroll
  for (int i = 0; i < 4; ++i)
#pragma unroll
    for (int j = 0; j < 4; ++j)
#pragma unroll
      for (int r = 0; r < 8; ++r) acc[i][j][r] = 0.0f;

  const float* arow[4];
  const float* brow[4];
#pragma unroll
  for (int i = 0; i < 4; ++i) {
    arow[i] = A + (size_t)(m0 + i * 16 + l16) * KDIM;
    brow[i] = Wu + (size_t)(n0 + i * 16 + l16) * KDIM;
  }

  for (int kk = 0; kk < KDIM; kk += 4) {
    const int ka = kk + 2 * half;
    v2f a[4], b[4];
#pragma unroll
    for (int i = 0; i < 4; ++i) a[i] = *(const v2f*)(arow[i] + ka);
#pragma unroll
    for (int j = 0; j < 4; ++j) b[j] = *(const v2f*)(brow[j] + ka);
#pragma unroll
    for (int i = 0; i < 4; ++i)
#pragma unroll
      for (int j = 0; j < 4; ++j)
        acc[i][j] = WMMA_F32(a[i], b[j], acc[i][j]);
  }

#pragma unroll
  for (int j = 0; j < 4; ++j) {
    const float bias = bu[n0 + j * 16 + l16];
#pragma unroll
    for (int i = 0; i < 4; ++i)
#pragma unroll
      for (int r = 0; r < 8; ++r)
        out[(size_t)(m0 + i * 16 + r + 8 * half) * KDIM + n0 + j * 16 + l16] =
            acc[i][j][r] + bias;
  }
}

// ---------------------------------------------------------------------------
// Launch: inputs are (x, Wk, Wq, Wv, Wu, bu)  <-- note Wk before Wq!
// Workspace: q, k, v, o buffers = 4 * 8192 * 1024 floats = 134 MB.
// ---------------------------------------------------------------------------
extern "C" void kernel_launch(void* const* d_in, const int* in_sizes, int n_in,
                              void* d_out, int out_size, void* d_ws, size_t ws_size,
                              hipStream_t stream) {
  const float* x  = (const float*)d_in[0];
  const float* Wk = (const float*)d_in[1];
  const float* Wq = (const float*)d_in[2];
  const float* Wv = (const float*)d_in[3];
  const float* Wu = (const float*)d_in[4];
  const float* bu = (const float*)d_in[5];
  float* out = (float*)d_out;

  const size_t MAT = (size_t)ROWS * KDIM;   // 8192*1024
  float* qb = (float*)d_ws;
  float* kb = qb + MAT;
  float* vb = kb + MAT;
  float* ob = vb + MAT;

  // 1) Q/K/V projections: 128 row-tiles x 16 col-tiles x 3 matrices, 1 wave each
  dim3 g1(ROWS / 64, KDIM / 64, 3);
  qkv_proj_kernel<<<g1, 32, 0, stream>>>(x, Wq, Wk, Wv, qb, kb, vb);

  // 2) Flash attention: 64 query tiles (32 rows each) x 16 heads x 4 batches
  dim3 g2(SEQ / 32, HEADS, BATCH);
  flash_attn_kernel<<<g2, 32, 0, stream>>>(qb, kb, vb, ob);

  // 3) Output projection + bias
  dim3 g3(ROWS / 64, KDIM / 64);
  out_proj_kernel<<<g3, 32, 0, stream>>>(ob, Wu, bu, out);
}